// WaveRNN_34857954574996
// MI455X (gfx1250) — compile-verified
//
#include <hip/hip_runtime.h>

#define NT 600
#define BATCH 2
#define H 256
#define W 256
#define NR 128
#define SLABS 4
#define ROWS_PER_WG 64          /* H / SLABS */
#define NWG (BATCH * SLABS)     /* 8 workgroups, one per WGP */
#define BLOCK 512               /* 16 wave32 waves */
#define LROWS (ROWS_PER_WG + 2) /* 66: slab + top/bottom halo rows */
#define LPITCH W
#define BUF_FLOATS (LROWS * LPITCH) /* 16896 floats */
#define DTDX 1.0e-4f                /* DT/DX = 0.001/10.0 */

typedef float v2f __attribute__((ext_vector_type(2)));
typedef float v8f __attribute__((ext_vector_type(8)));

__device__ __forceinline__ float Lmat(int i, int j) {
  int d = i - j;
  if (d == 0) return -2.0f;
  if (d == 1 || d == -1) return 1.0f;
  return 0.0f;
}

// Async DMA helpers (CDNA5 GLOBAL_*_ASYNC_*_LDS, tracked on ASYNCcnt).
// LDS flat pointers truncate to the wave-relative LDS byte address (the
// shared aperture base has zero low-32 bits).
__device__ __forceinline__ void async_lds_to_global(float* gdst,
                                                    const float* lsrc) {
  asm volatile("global_store_async_from_lds_b32 %0, %1, off" ::"v"(
                   (unsigned long long)(uintptr_t)gdst),
               "v"((unsigned)(uintptr_t)lsrc)
               : "memory");
}
__device__ __forceinline__ void async_global_to_lds(float* ldst,
                                                    const float* gsrc) {
  asm volatile("global_load_async_to_lds_b32 %0, %1, off" ::"v"(
                   (unsigned)(uintptr_t)ldst),
               "v"((unsigned long long)(uintptr_t)gsrc)
               : "memory");
}
__device__ __forceinline__ void wait_asynccnt0() {
  asm volatile("s_wait_asynccnt 0x0" ::: "memory");
}

__global__ __launch_bounds__(BLOCK) void wave_fdtd_persistent(
    const float* __restrict__ x, const float* __restrict__ vp,
    const int* __restrict__ sy, const int* __restrict__ sx,
    const int* __restrict__ ry, const int* __restrict__ rx,
    float* __restrict__ out, float* __restrict__ ws) {
  extern __shared__ float smem[]; // 2 * BUF_FLOATS floats (two time levels)

  const int wg = blockIdx.x;
  const int b = wg / SLABS; // batch
  const int s = wg % SLABS; // slab index within batch
  const int tid = threadIdx.x;
  const int rowBase = s * ROWS_PER_WG;

  unsigned* cnt = (unsigned*)ws;  // grid-barrier counter (zeroed by host)
  float* scrUp = ws + 16;         // [NWG][W]: each WG's first interior row
  float* scrDn = scrUp + NWG * W; // [NWG][W]: each WG's last interior row

  // Zero both LDS buffers (covers physical-edge halo rows = zero padding).
  for (int i = tid; i < 2 * BUF_FLOATS; i += BLOCK) smem[i] = 0.0f;

  // Receiver ownership: thread tid (<NR) owns receiver tid if in our slab.
  int recRow = -1, recCol = 0;
  if (tid < NR) {
    int yy = ry[tid], xx = rx[tid];
    if (yy >= rowBase && yy < rowBase + ROWS_PER_WG) {
      recRow = yy - rowBase + 1;
      recCol = xx;
    }
  }
  // Source ownership (handled by thread 0 of the owning WG).
  int srcOff = -1;
  {
    int yy = sy[b], xx = sx[b];
    if (yy >= rowBase && yy < rowBase + ROWS_PER_WG)
      srcOff = (yy - rowBase + 1) * LPITCH + xx;
  }

  // Wave tiling: 4 tile-rows x 16 tile-cols of 16x16 tiles; 4 tiles/wave.
  const int wv = tid >> 5;
  const int lane = tid & 31;
  const int hi = lane >> 4; // half-wave select (WMMA 32-bit layouts)
  const int lo = lane & 15;
  const int tr = wv >> 2;       // tile row 0..3
  const int tc0 = (wv & 3) * 4; // first of 4 tile cols

  // Constant stencil operands.
  // A-layout 16x4 f32: lane -> M = lo, VGPR j -> K = j + 2*hi
  // B-layout 4x16 f32: lane -> N = lo, VGPR j -> K = j + 2*hi
  v2f LvA[4], LhB[4], EvA, EhB;
#pragma unroll
  for (int k = 0; k < 4; ++k) {
#pragma unroll
    for (int j = 0; j < 2; ++j) {
      int kk = 4 * k + j + 2 * hi;
      LvA[k][j] = Lmat(lo, kk); // columns of Lv
      LhB[k][j] = Lmat(kk, lo); // rows of Lh
    }
  }
#pragma unroll
  for (int j = 0; j < 2; ++j) {
    int kk = j + 2 * hi;
    EvA[j] = ((lo == 0 && kk == 0) || (lo == 15 && kk == 1)) ? 1.0f : 0.0f;
    EhB[j] = ((kk == 0 && lo == 0) || (kk == 1 && lo == 15)) ? 1.0f : 0.0f;
  }

  // Pre-square c2 = (vp*DT/DX)^2 for our 4 tiles (registers, C/D layout).
  v8f c2t[4];
#pragma unroll
  for (int ti = 0; ti < 4; ++ti) {
    int tc = tc0 + ti;
#pragma unroll
    for (int r = 0; r < 8; ++r) {
      int gm = rowBase + tr * 16 + r + 8 * hi;
      int gn = tc * 16 + lo;
      float v = vp[gm * W + gn] * DTDX;
      c2t[ti][r] = v * v;
    }
  }

  __syncthreads();

  int cur = 0;
  for (int t = 0; t < NT; ++t) {
    float* u1 = smem + cur * BUF_FLOATS;       // current field (valid halos)
    float* u2 = smem + (cur ^ 1) * BUF_FLOATS; // previous field -> u_new

#pragma unroll
    for (int ti = 0; ti < 4; ++ti) {
      const int tc = tc0 + ti;
      const int r0 = 1 + tr * 16; // LDS row of tile row 0 (row 0 is halo)
      const int c0 = tc * 16;

      // Gather WMMA operands of u1 tile from LDS.
      v2f uB[4], uA[4];
#pragma unroll
      for (int k = 0; k < 4; ++k) {
#pragma unroll
        for (int j = 0; j < 2; ++j) {
          uB[k][j] = u1[(r0 + 4 * k + j + 2 * hi) * LPITCH + (c0 + lo)];
          uA[k][j] = u1[(r0 + lo) * LPITCH + (c0 + 4 * k + j + 2 * hi)];
        }
      }
      // Halo operands: unconditional in-bounds loads + v_cndmask select
      // (avoids exec-mask save/restore around predicated ds_loads).
      float tpv = u1[(r0 - 1) * LPITCH + (c0 + lo)];
      float btv = u1[(r0 + 16) * LPITCH + (c0 + lo)];
      float lfv = u1[(r0 + lo) * LPITCH + (c0 - 1)];  // min offset 255: in-bounds
      float rtv = u1[(r0 + lo) * LPITCH + (c0 + 16)]; // max offset 16640: in-bounds
      v2f hB, hA;
      hB[0] = (hi == 0) ? tpv : 0.0f;
      hB[1] = (hi == 0) ? btv : 0.0f;
      hA[0] = (hi == 0 && c0 > 0) ? lfv : 0.0f;
      hA[1] = (hi == 0 && c0 + 16 < W) ? rtv : 0.0f;

      // Lap(U) = Lv*U + U*Lh + e0(x)top + e15(x)bot + left(x)e0' + right(x)e15'
      v8f acc = {};
#pragma unroll
      for (int k = 0; k < 4; ++k)
        acc = __builtin_amdgcn_wmma_f32_16x16x4_f32(false, LvA[k], false, uB[k],
                                                    (short)0, acc, false, false);
#pragma unroll
      for (int k = 0; k < 4; ++k)
        acc = __builtin_amdgcn_wmma_f32_16x16x4_f32(false, uA[k], false, LhB[k],
                                                    (short)0, acc, false, false);
      acc = __builtin_amdgcn_wmma_f32_16x16x4_f32(false, EvA, false, hB,
                                                  (short)0, acc, false, false);
      acc = __builtin_amdgcn_wmma_f32_16x16x4_f32(false, hA, false, EhB,
                                                  (short)0, acc, false, false);

      // u_new = 2*u1 - u2 + c2 .* Lap(u1); in-place over the u2 buffer.
#pragma unroll
      for (int r = 0; r < 8; ++r) {
        int off = (r0 + r + 8 * hi) * LPITCH + (c0 + lo);
        float a1 = u1[off];
        float a2 = u2[off];
        u2[off] = 2.0f * a1 - a2 + c2t[ti][r] * acc[r];
      }
    }
    __syncthreads();

    // Source scatter (matches reference: add, then gather).
    if (tid == 0 && srcOff >= 0) u2[srcOff] += x[t * BATCH + b];
    __syncthreads();

    // Receiver gather into out[t, b, r].
    if (recRow >= 0)
      out[(size_t)t * (BATCH * NR) + b * NR + tid] =
          u2[recRow * LPITCH + recCol];

    // Halo export: LDS -> global scratch via async DMA (ASYNCcnt path).
    if (tid < W) {
      if (s > 0)
        async_lds_to_global(&scrUp[wg * W + tid], &u2[1 * LPITCH + tid]);
      if (s < SLABS - 1)
        async_lds_to_global(&scrDn[wg * W + tid],
                            &u2[ROWS_PER_WG * LPITCH + tid]);
    }
    wait_asynccnt0();

    // Device-wide barrier (monotonic counter; release/acquire, agent scope).
    __builtin_amdgcn_fence(__ATOMIC_RELEASE, "agent");
    __syncthreads();
    if (tid == 0) {
      __hip_atomic_fetch_add(cnt, 1u, __ATOMIC_RELEASE,
                             __HIP_MEMORY_SCOPE_AGENT);
      unsigned target = (unsigned)(t + 1) * NWG;
      while (__hip_atomic_load(cnt, __ATOMIC_ACQUIRE,
                               __HIP_MEMORY_SCOPE_AGENT) < target)
        __builtin_amdgcn_s_sleep(2);
    }
    __syncthreads();
    __builtin_amdgcn_fence(__ATOMIC_ACQUIRE, "agent");

    // Halo import: global scratch -> LDS halo rows via async DMA.
    if (tid < W) {
      if (s > 0)
        async_global_to_lds(&u2[0 * LPITCH + tid], &scrDn[(wg - 1) * W + tid]);
      if (s < SLABS - 1)
        async_global_to_lds(&u2[(ROWS_PER_WG + 1) * LPITCH + tid],
                            &scrUp[(wg + 1) * W + tid]);
    }
    wait_asynccnt0();
    __syncthreads();

    cur ^= 1;
  }
}

extern "C" void kernel_launch(void* const* d_in, const int* in_sizes, int n_in,
                              void* d_out, int out_size, void* d_ws,
                              size_t ws_size, hipStream_t stream) {
  (void)in_sizes;
  (void)n_in;
  (void)out_size;
  (void)ws_size;
  const float* x = (const float*)d_in[0];
  const float* vp = (const float*)d_in[1];
  const int* src_y = (const int*)d_in[2];
  const int* src_x = (const int*)d_in[3];
  const int* rec_y = (const int*)d_in[4];
  const int* rec_x = (const int*)d_in[5];
  float* out = (float*)d_out;
  float* ws = (float*)d_ws;

  // Allow >64KB dynamic LDS (we use 135168 B; WGP has 320 KB).
  const int shmem = 2 * BUF_FLOATS * (int)sizeof(float);
  hipFuncSetAttribute((const void*)wave_fdtd_persistent,
                      hipFuncAttributeMaxDynamicSharedMemorySize, shmem);

  // Reset grid-barrier counter (graph-capture-safe stream op).
  hipMemsetAsync(d_ws, 0, 64, stream);

  wave_fdtd_persistent<<<dim3(NWG), dim3(BLOCK), shmem, stream>>>(
      x, vp, src_y, src_x, rec_y, rec_x, out, ws);
}